// STU_9002251452587
// MI455X (gfx1250) — compile-verified
//
#include <hip/hip_runtime.h>

// ============================================================================
// STU forward for MI455X (gfx1250).
//
// Pipeline (FFT length n = 8192, rfft bins NF = 4097):
//   K1: 296 real-input FFTs -> v_f[k][f], U_f[b][s][d][f]   (complex, f32)
//   K2: freq-domain GEMM with v_wmma_f32_16x16x4_f32:
//         spec_f[b][s][f][o] = sum_{k,d} v_f[f,k] * U_f[b,s,f,d] * M_s[k,d,o]
//       rows = (re,im) x frequencies, K-dim = 40*64 = 2560, N = 64.
//       k-outer / d-inner: per-lane (p,q) from v_f once per k, A elem = 2 FMA;
//       one A fragment feeds 4 WMMAs; B chunk staged in LDS per k.
//   K3: per (b,o): two inverse FFTs (plus/minus spectra), (-1)^l on the minus
//       branch, fused 3-tap AR term, store strided by DOUT.
// ============================================================================

#define LSEQ   4096
#define NFFT   8192
#define LOG2N  13
#define NF     4097          // rfft bins
#define KFILT  40
#define DIN    64
#define DOUT   64
#define BSZ    2
#define FREQ   64            // frequencies per K2 block (128 GEMM rows)
#define LBS    65            // padded LDS row stride for B chunk (bank-conflict free)

typedef __attribute__((ext_vector_type(2))) float v2f;
typedef __attribute__((ext_vector_type(8))) float v8f;

static __device__ __forceinline__ v8f wmma4(v2f a, v2f b, v8f c) {
  // 8-arg pattern: (neg_a, A, neg_b, B, c_mod, C, reuse_a, reuse_b)
  return __builtin_amdgcn_wmma_f32_16x16x4_f32(false, a, false, b,
                                               (short)0, c, false, false);
}

// In-place radix-2 DIT complex FFT over LDS, 8192 points, 256 threads.
// Input must be pre-permuted to bit-reversed order. sgn=-1 fwd, +1 inv.
static __device__ __forceinline__ void fft8192(float2* s, int tid, float sgn) {
  const float TWO_PI = 6.28318530717958647692f;
#pragma unroll 1
  for (int st = 1; st <= LOG2N; ++st) {
    const int half = 1 << (st - 1);
    const float base = sgn * TWO_PI / (float)(1 << st);
    for (int t = tid; t < (NFFT >> 1); t += 256) {
      const int pos = t & (half - 1);
      const int i0 = ((t >> (st - 1)) << st) | pos;
      const int i1 = i0 + half;
      float sw, cw;
      __sincosf(base * (float)pos, &sw, &cw);
      const float2 a = s[i0];
      const float2 b = s[i1];
      const float tr = cw * b.x - sw * b.y;
      const float ti = cw * b.y + sw * b.x;
      s[i0].x = a.x + tr; s[i0].y = a.y + ti;
      s[i1].x = a.x - tr; s[i1].y = a.y - ti;
    }
    __syncthreads();
  }
}

// ---------------------------------------------------------------------------
// Kernel 1: forward FFTs of phi columns and of x / x*sgn channels.
//   blocks 0..39   : phi[:,k]             -> ws_vf[k][f]
//   blocks 40..295 : x[b,:,d] (*sgn if s) -> ws_uf[(b*2+s)*64+d][f]
// ---------------------------------------------------------------------------
extern "C" __global__ void __launch_bounds__(256)
stu_fft_fwd(const float* __restrict__ x, const float* __restrict__ phi,
            float* __restrict__ ws_vf, float* __restrict__ ws_uf) {
  extern __shared__ char smem_raw[];
  float2* s = (float2*)smem_raw;
  const int tid = threadIdx.x;
  const int blk = blockIdx.x;

  if (blk < KFILT) {
    const int k = blk;
    for (int i = tid; i < NFFT; i += 256) {
      const float v = (i < LSEQ) ? phi[(size_t)i * KFILT + k] : 0.0f;
      const int r = __brev((unsigned)i) >> (32 - LOG2N);
      s[r].x = v; s[r].y = 0.0f;
    }
    __syncthreads();
    fft8192(s, tid, -1.0f);
    float2* out = (float2*)ws_vf + (size_t)k * NF;
    for (int f = tid; f < NF; f += 256) out[f] = s[f];
  } else {
    const int idx = blk - KFILT;   // = bs*64 + d
    const int d  = idx & 63;
    const int bs = idx >> 6;
    const int b  = bs >> 1;
    const int sg = bs & 1;
    for (int i = tid; i < NFFT; i += 256) {
      float v = 0.0f;
      if (i < LSEQ) {
        v = x[((size_t)(b * LSEQ + i)) * DIN + d];
        if (sg && (i & 1)) v = -v;
      }
      const int r = __brev((unsigned)i) >> (32 - LOG2N);
      s[r].x = v; s[r].y = 0.0f;
    }
    __syncthreads();
    fft8192(s, tid, -1.0f);
    float2* out = (float2*)ws_uf + (size_t)idx * NF;
    for (int f = tid; f < NF; f += 256) out[f] = s[f];
  }
}

// ---------------------------------------------------------------------------
// Kernel 2: frequency-domain contraction via WMMA f32 16x16x4.
// grid.x = 65 frequency tiles (64 freqs each), grid.y = bs in [0,4).
// GEMM rows are (re,im)-interleaved frequencies: row = 2*f_local + part.
// A[row, kd] = part==0 ? Re(v[f,k]*U[f,d]) : Im(v[f,k]*U[f,d])
//            = p*U.re + q*U.im   with (p,q) = part ? (v.im, v.re) : (v.re,-v.im)
// B[kd, o]   = M_s[k,d,o]  (M_phi_plus for s=0, M_phi_minus for s=1).
// Each of 8 waves owns one 16-row tile x four 16-col tiles (full N=64).
// Output planes: spec_re/spec_im [bs][f][o] (o contiguous -> coalesced).
// ---------------------------------------------------------------------------
extern "C" __global__ void __launch_bounds__(256)
stu_freq_gemm(const float* __restrict__ ws_vf, const float* __restrict__ ws_uf,
              const float* __restrict__ Mp, const float* __restrict__ Mm,
              float* __restrict__ spec_re, float* __restrict__ spec_im) {
  extern __shared__ char smem_raw[];
  float2* lv = (float2*)smem_raw;        // [KFILT][FREQ]
  float2* lu = lv + KFILT * FREQ;        // [DIN][FREQ]
  float*  lB = (float*)(lu + DIN * FREQ);// [DIN][LBS]  (padded)
  const int tid = threadIdx.x;
  const int bs  = blockIdx.y;
  const int f0  = blockIdx.x * FREQ;

  const float2* vf2 = (const float2*)ws_vf;
  const float2* uf2 = (const float2*)ws_uf;
  for (int i = tid; i < KFILT * FREQ; i += 256) {
    const int k = i >> 6, fl = i & (FREQ - 1), f = f0 + fl;
    float2 v; v.x = 0.0f; v.y = 0.0f;
    if (f < NF) v = vf2[(size_t)k * NF + f];
    lv[i] = v;
  }
  for (int i = tid; i < DIN * FREQ; i += 256) {
    const int d = i >> 6, fl = i & (FREQ - 1), f = f0 + fl;
    float2 v; v.x = 0.0f; v.y = 0.0f;
    if (f < NF) v = uf2[((size_t)bs * DIN + d) * NF + f];
    lu[i] = v;
  }

  const float* Bmat = (bs & 1) ? Mm : Mp;   // [k][d][o], row-major
  const int lane = tid & 31;
  const int m    = lane & 15;   // WMMA M index (A) / N index (B,C)
  const int g    = lane >> 4;   // selects K pair {2g, 2g+1}
  const int rt   = tid >> 5;    // wave id == row tile (8 x 16 rows = 128)
  const int row  = rt * 16 + m;
  const int fl   = row >> 1;    // local frequency
  const int part = row & 1;     // 0 = Re, 1 = Im
  const float2* luf = lu + fl;

  v8f c[4];
#pragma unroll
  for (int ct = 0; ct < 4; ++ct) c[ct] = v8f{0.f,0.f,0.f,0.f,0.f,0.f,0.f,0.f};

#pragma unroll 1
  for (int k = 0; k < KFILT; ++k) {
    __syncthreads();   // previous B chunk fully consumed (covers lv/lu on k=0)
    // stage B[k][64][64] -> LDS with padded stride
    for (int i = tid; i < DIN * DOUT; i += 256)
      lB[(i >> 6) * LBS + (i & 63)] = Bmat[(size_t)k * DIN * DOUT + i];
    __syncthreads();

    const float2 vv = lv[k * FREQ + fl];
    const float p = part ? vv.y : vv.x;
    const float q = part ? vv.x : -vv.y;

#pragma unroll 4
    for (int d0 = 0; d0 < DIN; d0 += 4) {
      const int d = d0 + 2 * g;           // even; d+1 stays inside this k
      const float2 u0 = luf[d * FREQ];
      const float2 u1 = luf[(d + 1) * FREQ];
      v2f a;
      a[0] = p * u0.x + q * u0.y;
      a[1] = p * u1.x + q * u1.y;
      const float* b0 = lB + d * LBS + m;
      const float* b1 = lB + (d + 1) * LBS + m;
#pragma unroll
      for (int ct = 0; ct < 4; ++ct) {
        v2f bf;
        bf[0] = b0[ct * 16];
        bf[1] = b1[ct * 16];
        c[ct] = wmma4(a, bf, c[ct]);
      }
    }
  }

#pragma unroll
  for (int ct = 0; ct < 4; ++ct) {
#pragma unroll
    for (int i = 0; i < 8; ++i) {
      const int rl = i + 8 * g;           // C layout: VGPR i -> M = i + 8*(lane/16)
      const int r  = rt * 16 + rl;
      const int f  = f0 + (r >> 1);
      if (f < NF) {
        float* plane = (r & 1) ? spec_im : spec_re;
        plane[((size_t)bs * NF + f) * DOUT + ct * 16 + m] = c[ct][i];
      }
    }
  }
}

// ---------------------------------------------------------------------------
// Kernel 3: per (b,o): two inverse FFTs (plus/minus), combine with (-1)^l on
// the minus branch, fuse 3-tap AR term, write out[b,l,o].
// ---------------------------------------------------------------------------
extern "C" __global__ void __launch_bounds__(256)
stu_ifft_ar(const float* __restrict__ spec_re, const float* __restrict__ spec_im,
            const float* __restrict__ x, const float* __restrict__ Mar,
            float* __restrict__ out) {
  extern __shared__ char smem_raw[];
  float2* s   = (float2*)smem_raw;
  float* acc  = (float*)(smem_raw + (size_t)NFFT * sizeof(float2));
  float* mrow = acc + LSEQ;
  const int tid = threadIdx.x;
  const int b = blockIdx.x >> 6;
  const int o = blockIdx.x & 63;
  const float scale = 1.0f / (float)NFFT;

  for (int sg = 0; sg < 2; ++sg) {
    const int bs = b * 2 + sg;
    // rebuild full conjugate-symmetric spectrum, bit-reversed into LDS
    for (int i = tid; i < NFFT; i += 256) {
      const int f = (i <= LSEQ) ? i : (NFFT - i);
      float re = spec_re[((size_t)bs * NF + f) * DOUT + o];
      float im = spec_im[((size_t)bs * NF + f) * DOUT + o];
      if (i > LSEQ) im = -im;
      const int r = __brev((unsigned)i) >> (32 - LOG2N);
      s[r].x = re; s[r].y = im;
    }
    __syncthreads();
    fft8192(s, tid, +1.0f);
    for (int l = tid; l < LSEQ; l += 256) {
      const float v = s[l].x * scale;
      if (sg == 0) acc[l] = v;
      else         acc[l] += (l & 1) ? -v : v;   // U_minus carries sgn[l]
    }
    __syncthreads();
  }

  // AR weights for this output channel: M[o, d, i], (d,i) row-major
  for (int i = tid; i < DIN * 3; i += 256) mrow[i] = Mar[(size_t)o * (DIN * 3) + i];
  __syncthreads();

  for (int l = tid; l < LSEQ; l += 256) {
    float a = acc[l];
#pragma unroll
    for (int i = 0; i < 3; ++i) {
      if (l >= i) {
        const float* xr = x + ((size_t)(b * LSEQ + (l - i))) * DIN;
        float dv = 0.0f;
#pragma unroll 8
        for (int d = 0; d < DIN; ++d) dv += xr[d] * mrow[d * 3 + i];
        a += dv;
      }
    }
    out[((size_t)(b * LSEQ + l)) * DOUT + o] = a;
  }
}

// ---------------------------------------------------------------------------
// Host launcher.  Workspace layout (floats):
//   ws_vf  : 40*4097*2            =   327,760
//   ws_uf  : 4*64*4097*2          = 2,097,664
//   spec_re: 4*4097*64            = 1,048,832
//   spec_im: 4*4097*64            = 1,048,832    total ~18.1 MB
// ---------------------------------------------------------------------------
extern "C" void kernel_launch(void* const* d_in, const int* in_sizes, int n_in,
                              void* d_out, int out_size, void* d_ws, size_t ws_size,
                              hipStream_t stream) {
  (void)in_sizes; (void)n_in; (void)out_size; (void)ws_size;
  const float* x   = (const float*)d_in[0];  // (2,4096,64)
  const float* phi = (const float*)d_in[1];  // (4096,40)
  const float* Mar = (const float*)d_in[2];  // (64,64,3)
  const float* Mp  = (const float*)d_in[3];  // (40,64,64)
  const float* Mm  = (const float*)d_in[4];  // (40,64,64)
  float* out = (float*)d_out;
  float* ws  = (float*)d_ws;

  const size_t vf_off  = 0;
  const size_t uf_off  = vf_off  + (size_t)KFILT * NF * 2;
  const size_t sre_off = uf_off  + (size_t)BSZ * 2 * DIN * NF * 2;
  const size_t sim_off = sre_off + (size_t)BSZ * 2 * NF * DOUT;
  float* ws_vf = ws + vf_off;
  float* ws_uf = ws + uf_off;
  float* ws_sr = ws + sre_off;
  float* ws_si = ws + sim_off;

  stu_fft_fwd<<<dim3(KFILT + BSZ * 2 * DIN), dim3(256),
                (size_t)NFFT * sizeof(float2), stream>>>(x, phi, ws_vf, ws_uf);

  const size_t k2_lds = (size_t)(KFILT + DIN) * FREQ * sizeof(float2)
                      + (size_t)DIN * LBS * sizeof(float);
  stu_freq_gemm<<<dim3((NF + FREQ - 1) / FREQ, BSZ * 2), dim3(256), k2_lds, stream>>>(
      ws_vf, ws_uf, Mp, Mm, ws_sr, ws_si);

  stu_ifft_ar<<<dim3(BSZ * DOUT), dim3(256),
                (size_t)NFFT * sizeof(float2) + (size_t)(LSEQ + DIN * 3) * sizeof(float),
                stream>>>(ws_sr, ws_si, x, Mar, out);
}